// Controller_90941637526138
// MI455X (gfx1250) — compile-verified
//
#include <hip/hip_runtime.h>
#include <hip/hip_bf16.h>

typedef __attribute__((ext_vector_type(2))) float v2f;
typedef __attribute__((ext_vector_type(8))) float v8f;

#define BB 8
#define TT 512
#define CC 32
#define EPSV 1e-8f

// ---------------------------------------------------------------------------
// Pass 1: sequential scan over T. One wave per batch, one lane per channel.
// Produces E1 = exp(x1 - m1[t]), E2 = exp(x2 - m2[t]) and the row maxes.
// ---------------------------------------------------------------------------
__global__ __launch_bounds__(256) void ctrl_prep_kernel(
    const float* __restrict__ coref, const float* __restrict__ overwrite,
    float* __restrict__ e1, float* __restrict__ e2,
    float* __restrict__ m1a, float* __restrict__ m2a)
{
    const int tid = threadIdx.x;
    const int b   = tid >> 5;   // wave index == batch
    const int c   = tid & 31;   // lane == channel

    float cum = 0.0f;
    for (int t = 0; t < TT; ++t) {
        const size_t idx = ((size_t)b * TT + t) * CC + c;
        const float cor = coref[idx];
        const float ovr = overwrite[idx];
        cum += logf((1.0f - ovr) * (1.0f - EPSV) + EPSV);   // inclusive cumsum
        const float x1 = logf(cor + ovr) - cum;             // t1 contribution
        const float x2 = logf(cor) + cum;                   // t2 contribution

        // wave32 max-reduction across the 32 channels
        float mx1 = x1, mx2 = x2;
        #pragma unroll
        for (int m = 16; m >= 1; m >>= 1) {
            mx1 = fmaxf(mx1, __shfl_xor(mx1, m, 32));
            mx2 = fmaxf(mx2, __shfl_xor(mx2, m, 32));
        }

        e1[idx] = expf(x1 - mx1);
        e2[idx] = expf(x2 - mx2);
        if (c == 0) {
            m1a[b * TT + t] = mx1;
            m2a[b * TT + t] = mx2;
        }
    }
}

// ---------------------------------------------------------------------------
// Pass 2: per batch, S = E1 (512x32) * E2^T (32x512) via V_WMMA_F32_16X16X4_F32,
// then out = (t2 > t1) ? log(S) + m1[t1] + m2[t2] : 0.
// Block = 4 waves computing a 64x64 output tile; wave w owns rows [16w,16w+16).
// ---------------------------------------------------------------------------
__global__ __launch_bounds__(128) void ctrl_wmma_kernel(
    const float* __restrict__ e1, const float* __restrict__ e2,
    const float* __restrict__ m1a, const float* __restrict__ m2a,
    float* __restrict__ out)
{
    __shared__ float lE1[64][33];   // padded to dodge bank conflicts
    __shared__ float lE2[64][33];

    const int b    = blockIdx.z;
    const int row0 = blockIdx.y * 64;   // t1 tile base
    const int col0 = blockIdx.x * 64;   // t2 tile base
    const int tid  = threadIdx.x;
    const int lane = tid & 31;
    const int wave = tid >> 5;

    float* __restrict__ outb = out + (size_t)b * TT * TT;

    // Tile entirely at-or-below the diagonal: masked to zero, skip the GEMM.
    // (block-uniform branch; EXEC stays all-ones on the WMMA path)
    if (col0 < row0) {
        for (int i = tid; i < 64 * 64; i += 128) {
            const int r = i >> 6, cc = i & 63;
            outb[(size_t)(row0 + r) * TT + (col0 + cc)] = 0.0f;
        }
        return;
    }

    // Stage contiguous 64x32 fp32 tiles of E1/E2 into LDS.
    const float* __restrict__ g1 = e1 + ((size_t)b * TT + row0) * CC;
    const float* __restrict__ g2 = e2 + ((size_t)b * TT + col0) * CC;
    for (int i = tid; i < 64 * 32; i += 128) {
        const int r = i >> 5, cc = i & 31;
        lE1[r][cc] = g1[i];
        lE2[r][cc] = g2[i];
    }
    __syncthreads();

    const int half = lane >> 4;          // K lane-half per ISA 16x4 layout
    const int l16  = lane & 15;
    const int rloc = wave * 16 + l16;    // A-matrix row (M) for this lane

    #pragma unroll
    for (int nt = 0; nt < 4; ++nt) {     // four 16-wide column tiles
        v8f acc = {};
        const int cloc = nt * 16 + l16;  // B-matrix column (N) for this lane
        #pragma unroll
        for (int kk = 0; kk < CC; kk += 4) {
            const int k0 = kk + half * 2;
            v2f a, bm;
            a.x  = lE1[rloc][k0];
            a.y  = lE1[rloc][k0 + 1];
            bm.x = lE2[cloc][k0];
            bm.y = lE2[cloc][k0 + 1];
            acc = __builtin_amdgcn_wmma_f32_16x16x4_f32(
                /*neg_a=*/false, a, /*neg_b=*/false, bm,
                /*c_mod=*/(short)0, acc, /*reuse_a=*/false, /*reuse_b=*/false);
        }

        // Epilogue: log + row/col maxes + strict-upper-triangle mask.
        const int   t2  = col0 + cloc;
        const float mm2 = m2a[b * TT + t2];
        #pragma unroll
        for (int v = 0; v < 8; ++v) {
            const int t1 = row0 + wave * 16 + v + half * 8;  // C/D layout: M = v + 8*half
            float val = 0.0f;
            if (t2 > t1) val = logf(acc[v]) + m1a[b * TT + t1] + mm2;
            outb[(size_t)t1 * TT + t2] = val;
        }
    }
}

extern "C" void kernel_launch(void* const* d_in, const int* in_sizes, int n_in,
                              void* d_out, int out_size, void* d_ws, size_t ws_size,
                              hipStream_t stream) {
    const float* coref     = (const float*)d_in[0];
    const float* overwrite = (const float*)d_in[1];
    float* out = (float*)d_out;

    // workspace carve-up: E1, E2 [B*T*C] + m1, m2 [B*T]
    float* ws = (float*)d_ws;
    float* e1  = ws;
    float* e2  = e1 + (size_t)BB * TT * CC;
    float* m1a = e2 + (size_t)BB * TT * CC;
    float* m2a = m1a + (size_t)BB * TT;

    ctrl_prep_kernel<<<1, 256, 0, stream>>>(coref, overwrite, e1, e2, m1a, m2a);

    dim3 grid(TT / 64, TT / 64, BB);   // (t2 tiles, t1 tiles, batch)
    ctrl_wmma_kernel<<<grid, 128, 0, stream>>>(e1, e2, m1a, m2a, out);
}